// DAGCN_55190329753905
// MI455X (gfx1250) — compile-verified
//
#include <hip/hip_runtime.h>

typedef __attribute__((ext_vector_type(2))) float v2f;
typedef __attribute__((ext_vector_type(8))) float v8f;

static constexpr int NNODES = 250000;
static constexpr int NEDGES = 4000000;
static constexpr int DIM    = 64;
static constexpr int WSTR   = 68;   // LDS stride (floats): 68 % 64 = 4 -> conflict-free b64 reads

// ---------------------------------------------------------------------------
// SPMM: one wave32 per edge. Lane L owns features [2L, 2L+1] of the 64-wide
// row: float2 gather from h[col[e]] (rows are 256B-aligned -> coalesced
// global_load_b64), scale by val[e], scatter via global_atomic_add_f32.
// ---------------------------------------------------------------------------
__global__ __launch_bounds__(256) void spmm_kernel(
    const int*   __restrict__ rows,
    const int*   __restrict__ cols,
    const float* __restrict__ vals,
    const float* __restrict__ h,
    float*       __restrict__ out)
{
    long long tid = (long long)blockIdx.x * blockDim.x + threadIdx.x;
    int e = (int)(tid >> 5);
    if (e >= NEDGES) return;
    int lane = (int)(tid & 31);

    int   r = rows[e];
    int   c = cols[e];
    float v = vals[e];

    v2f hv = *(const v2f*)(h + (size_t)c * DIM + lane * 2);
    float* dst = out + (size_t)r * DIM + lane * 2;
    unsafeAtomicAdd(dst + 0, v * hv.x);   // global_atomic_add_f32
    unsafeAtomicAdd(dst + 1, v * hv.y);
}

// ---------------------------------------------------------------------------
// Dense layer via V_WMMA_F32_16X16X4_F32:  out = A @ W^T + b  (A: [N,64] f32)
// One wave per 16-row tile; 4 N-tiles of 16 cols; K=64 -> 16 WMMAs per tile.
// A frag: float2 at A[row0+m][k0+kb] (m=lane&15, kb=2*(lane>>4)).
// B = W^T -> frag is contiguous float2 at W[n][k0+kb], n = lane&15 (+16t),
// staged in LDS (stride 68) for bank-conflict-free reads.
// FINAL fuses the model epilogue: out = (x + h1 + (A@W^T+b)) / 3.
// ---------------------------------------------------------------------------
template<bool FINAL>
__global__ __launch_bounds__(256) void gemm_wmma_kernel(
    const float* __restrict__ A,
    const float* __restrict__ W,
    const float* __restrict__ bias,
    const float* __restrict__ x,
    const float* __restrict__ h1,
    float*       __restrict__ out)
{
    __shared__ float Wl[DIM * WSTR];
    __shared__ float bl[DIM];

    // Stage W (64x64 f32 = 16KB) into padded LDS, coalesced float2 copies.
    for (int i = threadIdx.x; i < DIM * (DIM / 2); i += 256) {
        int n  = i / (DIM / 2);
        int k2 = i % (DIM / 2);
        *(v2f*)(Wl + n * WSTR + k2 * 2) = *(const v2f*)(W + n * DIM + k2 * 2);
    }
    if (threadIdx.x < DIM) bl[threadIdx.x] = bias[threadIdx.x];
    __syncthreads();

    int wave = threadIdx.x >> 5;
    int lane = threadIdx.x & 31;
    int m    = lane & 15;
    int half = lane >> 4;
    int kb   = half * 2;
    int row0 = (blockIdx.x * 8 + wave) * 16;
    if (row0 >= NNODES) return;          // uniform per wave: EXEC stays all-1s for WMMA

    v8f acc[4] = {v8f{0.f}, v8f{0.f}, v8f{0.f}, v8f{0.f}};
    const float* Arow = A + (size_t)(row0 + m) * DIM;

#pragma unroll
    for (int k0 = 0; k0 < DIM; k0 += 4) {
        v2f a  = *(const v2f*)(Arow + k0 + kb);
        v2f b0 = *(const v2f*)(Wl + ( 0 + m) * WSTR + k0 + kb);
        v2f b1 = *(const v2f*)(Wl + (16 + m) * WSTR + k0 + kb);
        v2f b2 = *(const v2f*)(Wl + (32 + m) * WSTR + k0 + kb);
        v2f b3 = *(const v2f*)(Wl + (48 + m) * WSTR + k0 + kb);
        acc[0] = __builtin_amdgcn_wmma_f32_16x16x4_f32(false, a, false, b0, (short)0, acc[0], false, false);
        acc[1] = __builtin_amdgcn_wmma_f32_16x16x4_f32(false, a, false, b1, (short)0, acc[1], false, false);
        acc[2] = __builtin_amdgcn_wmma_f32_16x16x4_f32(false, a, false, b2, (short)0, acc[2], false, false);
        acc[3] = __builtin_amdgcn_wmma_f32_16x16x4_f32(false, a, false, b3, (short)0, acc[3], false, false);
    }

    // C/D layout: acc[t][v] = C[row0 + v + 8*half][16t + m]
#pragma unroll
    for (int t = 0; t < 4; ++t) {
        int n = t * 16 + m;
        float bv = bl[n];
#pragma unroll
        for (int v = 0; v < 8; ++v) {
            size_t idx = (size_t)(row0 + v + 8 * half) * DIM + n;
            float val = acc[t][v] + bv;
            if (FINAL)
                out[idx] = (x[idx] + h1[idx] + val) * (1.0f / 3.0f);
            else
                out[idx] = val;
        }
    }
}

// ---------------------------------------------------------------------------
// inputs: 0:x 1:edge_rows 2:edge_cols 3:edge_vals 4:W0 5:b0 6:W1 7:b1
// workspace: buf0 (64MB agg) | buf1 (64MB h1)   -> needs 128MB of d_ws
// ---------------------------------------------------------------------------
extern "C" void kernel_launch(void* const* d_in, const int* in_sizes, int n_in,
                              void* d_out, int out_size, void* d_ws, size_t ws_size,
                              hipStream_t stream) {
    const float* x    = (const float*)d_in[0];
    const int*   rows = (const int*)  d_in[1];
    const int*   cols = (const int*)  d_in[2];
    const float* vals = (const float*)d_in[3];
    const float* W0   = (const float*)d_in[4];
    const float* b0   = (const float*)d_in[5];
    const float* W1   = (const float*)d_in[6];
    const float* b1   = (const float*)d_in[7];
    float* out = (float*)d_out;

    const size_t feat_elems = (size_t)NNODES * DIM;
    const size_t feat_bytes = feat_elems * sizeof(float);
    float* buf0 = (float*)d_ws;              // aggregation target (atomics)
    float* buf1 = buf0 + feat_elems;         // h1

    const int spmm_blocks = (int)(((long long)NEDGES * 32 + 255) / 256);   // 500000
    const int gemm_blocks = (NNODES / 16 + 7) / 8;                         // 1954

    // Layer 1: agg = spmm(x); h1 = agg @ W0^T + b0
    hipMemsetAsync(buf0, 0, feat_bytes, stream);
    spmm_kernel<<<spmm_blocks, 256, 0, stream>>>(rows, cols, vals, x, buf0);
    gemm_wmma_kernel<false><<<gemm_blocks, 256, 0, stream>>>(
        buf0, W0, b0, nullptr, nullptr, buf1);

    // Layer 2: agg = spmm(h1); out = (x + h1 + (agg @ W1^T + b1)) / 3
    hipMemsetAsync(buf0, 0, feat_bytes, stream);
    spmm_kernel<<<spmm_blocks, 256, 0, stream>>>(rows, cols, vals, buf1, buf0);
    gemm_wmma_kernel<true><<<gemm_blocks, 256, 0, stream>>>(
        buf0, W1, b1, x, buf1, out);
}